// FinalGNN_38087769981223
// MI455X (gfx1250) — compile-verified
//
#include <hip/hip_runtime.h>
#include <math.h>
#include <stdint.h>

typedef __attribute__((ext_vector_type(2))) float v2f;
typedef __attribute__((ext_vector_type(8))) float v8f;

static constexpr int NB = 16;    // batch
static constexpr int NN = 256;   // nodes
static constexpr int ND = 64;    // feature dim

// ---------------------------------------------------------------------------
// Kernel 1: h = elu(x @ proj_w + proj_b); sq = rowwise ||h||^2;
//           a_i = h @ s_w1[:64]; a_j = h @ s_w1[64:]
// block = 256 threads = 4 rows x 64 cols
// ---------------------------------------------------------------------------
__global__ void proj_kernel(const float* __restrict__ x,
                            const float* __restrict__ proj_w,
                            const float* __restrict__ proj_b,
                            const float* __restrict__ s_w1,
                            float* __restrict__ h, float* __restrict__ sq,
                            float* __restrict__ a_i, float* __restrict__ a_j) {
  __shared__ float hl[4][ND];
  int t = threadIdx.x;
  int rl = t >> 6;          // row within block: 0..3
  int c  = t & 63;          // feature col
  int row = blockIdx.x * 4 + rl;       // 0..4095 (b*256 + n)

  float acc = proj_b[c];
#pragma unroll
  for (int k = 0; k < 4; ++k) acc += x[row * 4 + k] * proj_w[k * 64 + c];
  float hv = acc > 0.f ? acc : expf(acc) - 1.f;   // elu
  h[row * ND + c] = hv;
  hl[rl][c] = hv;
  __syncthreads();

  if (c == 0) {
    float s = 0.f;
    for (int k = 0; k < ND; ++k) s += hl[rl][k] * hl[rl][k];
    sq[row] = s;
  }

  float ai = 0.f, aj = 0.f;
  for (int k = 0; k < ND; ++k) {
    float hk = hl[rl][k];
    ai += hk * s_w1[k * 64 + c];
    aj += hk * s_w1[(64 + k) * 64 + c];
  }
  a_i[row * ND + c] = ai;
  a_j[row * ND + c] = aj;
}

// ---------------------------------------------------------------------------
// Kernel 2: per pair (i,j): pre = a_i[i]+a_j[j]+s_b1 -> LN -> ReLU -> dot s_w2
//           -> softplus -> store sinv = 1/(2*sigma^2 + 1e-6)
// grid = (jTile=16, iTile=16, b=16); block = 256 = 16x16 pairs
// pre[] kept in VGPRs (wave32: plenty of registers) -> single LDS pass
// ---------------------------------------------------------------------------
__global__ void sigma_kernel(const float* __restrict__ a_i,
                             const float* __restrict__ a_j,
                             const float* __restrict__ s_b1,
                             const float* __restrict__ ln_g,
                             const float* __restrict__ ln_b,
                             const float* __restrict__ s_w2,
                             const float* __restrict__ s_b2,
                             float* __restrict__ sinv) {
  __shared__ float ail[16 * ND], ajl[16 * ND];
  __shared__ float sb[ND], lg[ND], lb[ND], w2[ND];
  int t = threadIdx.x;
  int j0 = blockIdx.x * 16, i0 = blockIdx.y * 16, b = blockIdx.z;
  int base_i = (b * NN + i0) * ND;
  int base_j = (b * NN + j0) * ND;
  for (int idx = t; idx < 16 * ND; idx += 256) {
    ail[idx] = a_i[base_i + idx];
    ajl[idx] = a_j[base_j + idx];
  }
  if (t < ND) { sb[t] = s_b1[t]; lg[t] = ln_g[t]; lb[t] = ln_b[t]; w2[t] = s_w2[t]; }
  __syncthreads();

  int ti = t >> 4, tj = t & 15;
  const float* pi = &ail[ti * ND];
  const float* pj = &ajl[tj * ND];

  float pre[ND];
  float sum = 0.f, sumsq = 0.f;
#pragma unroll
  for (int c = 0; c < ND; ++c) {
    float v = pi[c] + pj[c] + sb[c];
    pre[c] = v;
    sum += v; sumsq += v * v;
  }
  float mean = sum * (1.f / 64.f);
  float var  = sumsq * (1.f / 64.f) - mean * mean;
  float rstd = rsqrtf(var + 1e-5f);

  float acc = s_b2[0];
#pragma unroll
  for (int c = 0; c < ND; ++c) {
    float l = (pre[c] - mean) * rstd * lg[c] + lb[c];
    l = fmaxf(l, 0.f);
    acc += l * w2[c];
  }
  float sp = (acc > 20.f) ? acc : log1pf(expf(acc));   // softplus
  sinv[(b * NN + i0 + ti) * NN + j0 + tj] = 1.f / (2.f * sp * sp + 1e-6f);
}

// ---------------------------------------------------------------------------
// Kernel 3: gram tile via v_wmma_f32_16x16x4_f32, fused into
//           adj = exp(-max(sq_i+sq_j-2G,0) * sinv)
// grid = (jGroup=2, iTile=16, b=16); block = 256 (8 waves, one j-tile each)
// ---------------------------------------------------------------------------
__global__ void gram_adj_kernel(const float* __restrict__ h,
                                const float* __restrict__ sq,
                                const float* __restrict__ sinv,
                                float* __restrict__ adj) {
  int t = threadIdx.x, lane = t & 31, wave = t >> 5;
  int b = blockIdx.z, i0 = blockIdx.y * 16;
  int j0 = (blockIdx.x * 8 + wave) * 16;
  int half = lane >> 4;     // 0/1 : which K pair this half-wave holds
  int lm = lane & 15;       // M (A) / N (B) index

  const float* hb = h + b * NN * ND;
  const float* arow = hb + (i0 + lm) * ND + 2 * half;  // A: rows of h (i-tile)
  const float* brow = hb + (j0 + lm) * ND + 2 * half;  // B: rows of h (j-tile)

  v8f c = {};
#pragma unroll
  for (int k = 0; k < ND; k += 4) {
    v2f a  = *(const v2f*)(arow + k);
    v2f bf = *(const v2f*)(brow + k);
    c = __builtin_amdgcn_wmma_f32_16x16x4_f32(false, a, false, bf, (short)0, c,
                                              false, false);
  }

  int j = j0 + lm;
  float sqj = sq[b * NN + j];
#pragma unroll
  for (int v = 0; v < 8; ++v) {
    int i = i0 + v + 8 * half;                 // C layout: M = v + 8*(lane/16)
    float d2 = sq[b * NN + i] + sqj - 2.f * c[v];
    d2 = fmaxf(d2, 0.f);
    adj[(b * NN + i) * NN + j] = expf(-d2 * sinv[(b * NN + i) * NN + j]);
  }
}

// ---------------------------------------------------------------------------
// Kernel 4: tmp = adj @ h  ([256x256]@[256x64] per batch) via WMMA, K=256.
// Block <-> (b, d-tile). The shared B panel h[b, :, d0:d0+16] (16 KB) is
// staged into LDS once per block with GLOBAL_LOAD_ASYNC_TO_LDS_B128
// (ASYNCcnt path), then all 8 waves feed B fragments from LDS.
// Each wave computes TWO i-tiles per K step, sharing one B fragment.
// grid = 64 blocks x 256 threads.
// ---------------------------------------------------------------------------
__global__ void agg1_kernel(const float* __restrict__ adj,
                            const float* __restrict__ h,
                            float* __restrict__ tmp) {
  __shared__ float panel[NN * 16];            // [k][d] 256x16 floats = 16 KB
  int t = threadIdx.x, lane = t & 31, wave = t >> 5;
  int b  = blockIdx.x >> 2;                   // 0..15
  int d0 = (blockIdx.x & 3) * 16;             // 0/16/32/48

  // --- async stage: thread t copies row k=t (16 floats) via 4x B128 ---
  {
    const float* src = h + ((size_t)(b * NN + t)) * ND + d0;   // 16B aligned
    uint32_t ldsoff = (uint32_t)(uintptr_t)&panel[t * 16];     // LDS byte addr
#pragma unroll
    for (int q = 0; q < 4; ++q) {
      const float* g = src + 4 * q;
      uint32_t l = ldsoff + 16 * q;
      asm volatile("global_load_async_to_lds_b128 %0, %1, off"
                   :: "v"(l), "v"(g) : "memory");
    }
  }
  asm volatile("s_wait_asynccnt 0x0" ::: "memory");
  __syncthreads();

  int half = lane >> 4, lm = lane & 15;
  int i0a = (wave * 2) * 16;                  // i-tile pair for this wave
  int i0b = (wave * 2 + 1) * 16;
  const float* arow0 = adj + ((size_t)b * NN + i0a + lm) * NN + 2 * half;
  const float* arow1 = adj + ((size_t)b * NN + i0b + lm) * NN + 2 * half;

  v8f c0 = {}, c1 = {};
#pragma unroll 4
  for (int k = 0; k < NN; k += 4) {
    int kk = k + 2 * half;
    v2f bf;
    bf.x = panel[kk * 16 + lm];
    bf.y = panel[(kk + 1) * 16 + lm];
    v2f a0 = *(const v2f*)(arow0 + k);
    v2f a1 = *(const v2f*)(arow1 + k);
    c0 = __builtin_amdgcn_wmma_f32_16x16x4_f32(false, a0, false, bf, (short)0,
                                               c0, false, false);
    c1 = __builtin_amdgcn_wmma_f32_16x16x4_f32(false, a1, false, bf, (short)0,
                                               c1, false, false);
  }
#pragma unroll
  for (int v = 0; v < 8; ++v) {
    int ia = i0a + v + 8 * half;
    int ib = i0b + v + 8 * half;
    tmp[((size_t)b * NN + ia) * ND + d0 + lm] = c0[v];
    tmp[((size_t)b * NN + ib) * ND + d0 + lm] = c1[v];
  }
}

// ---------------------------------------------------------------------------
// Kernel 5: h_res = elu(tmp @ gnn_w + h) via WMMA, K=64
// ---------------------------------------------------------------------------
__global__ void agg2_kernel(const float* __restrict__ tmp,
                            const float* __restrict__ gnn_w,
                            const float* __restrict__ h,
                            float* __restrict__ h_res) {
  int t = threadIdx.x, lane = t & 31, wave = t >> 5;
  int w = blockIdx.x * 8 + wave;
  int d0 = (w & 3) * 16;
  int i0 = ((w >> 2) & 15) * 16;
  int b = w >> 6;
  int half = lane >> 4, lm = lane & 15;

  const float* arow = tmp + (b * NN + i0 + lm) * ND + 2 * half;

  v8f c = {};
#pragma unroll
  for (int k = 0; k < ND; k += 4) {
    v2f a = *(const v2f*)(arow + k);
    int kk = k + 2 * half;
    v2f bf;
    bf.x = gnn_w[kk * ND + d0 + lm];
    bf.y = gnn_w[(kk + 1) * ND + d0 + lm];
    c = __builtin_amdgcn_wmma_f32_16x16x4_f32(false, a, false, bf, (short)0, c,
                                              false, false);
  }
#pragma unroll
  for (int v = 0; v < 8; ++v) {
    int i = i0 + v + 8 * half;
    int idx = (b * NN + i) * ND + d0 + lm;
    float val = c[v] + h[idx];
    h_res[idx] = val > 0.f ? val : expf(val) - 1.f;   // elu
  }
}

// ---------------------------------------------------------------------------
// Kernel 6: mean/max pool over n, then classifier head. One block per batch.
// ---------------------------------------------------------------------------
__global__ void head_kernel(const float* __restrict__ h_res,
                            const float* __restrict__ c_w1,
                            const float* __restrict__ c_b1,
                            const float* __restrict__ bn_g,
                            const float* __restrict__ bn_b,
                            const float* __restrict__ bn_mean,
                            const float* __restrict__ bn_var,
                            const float* __restrict__ c_w2,
                            const float* __restrict__ c_b2,
                            float* __restrict__ out) {
  __shared__ float pooled[2 * ND], zl[ND];
  int t = threadIdx.x, b = blockIdx.x;
  if (t < ND) {
    float s = 0.f, m = -INFINITY;
    for (int r = 0; r < NN; ++r) {
      float v = h_res[(b * NN + r) * ND + t];
      s += v; m = fmaxf(m, v);
    }
    pooled[t] = s * (1.f / NN);
    pooled[ND + t] = m;
  }
  __syncthreads();
  if (t < ND) {
    float z = c_b1[t];
    for (int k = 0; k < 2 * ND; ++k) z += pooled[k] * c_w1[k * ND + t];
    z = (z - bn_mean[t]) * rsqrtf(bn_var[t] + 1e-5f) * bn_g[t] + bn_b[t];
    z = z > 0.f ? z : expf(z) - 1.f;   // elu (dropout = identity in eval)
    zl[t] = z;
  }
  __syncthreads();
  if (t < 2) {
    float o = c_b2[t];
    for (int k = 0; k < ND; ++k) o += zl[k] * c_w2[k * 2 + t];
    out[b * 2 + t] = o;
  }
}

// ---------------------------------------------------------------------------
extern "C" void kernel_launch(void* const* d_in, const int* in_sizes, int n_in,
                              void* d_out, int out_size, void* d_ws, size_t ws_size,
                              hipStream_t stream) {
  const float* x       = (const float*)d_in[0];
  const float* proj_w  = (const float*)d_in[1];
  const float* proj_b  = (const float*)d_in[2];
  const float* s_w1    = (const float*)d_in[3];
  const float* s_b1    = (const float*)d_in[4];
  const float* ln_g    = (const float*)d_in[5];
  const float* ln_b    = (const float*)d_in[6];
  const float* s_w2    = (const float*)d_in[7];
  const float* s_b2    = (const float*)d_in[8];
  const float* gnn_w   = (const float*)d_in[9];
  const float* c_w1    = (const float*)d_in[10];
  const float* c_b1    = (const float*)d_in[11];
  const float* bn_g    = (const float*)d_in[12];
  const float* bn_b    = (const float*)d_in[13];
  const float* bn_mean = (const float*)d_in[14];
  const float* bn_var  = (const float*)d_in[15];
  const float* c_w2    = (const float*)d_in[16];
  const float* c_b2    = (const float*)d_in[17];
  float* out = (float*)d_out;

  // workspace carve-up (floats)
  float* ws = (float*)d_ws;
  const size_t R = (size_t)NB * NN;           // 4096 rows
  float* h     = ws;                          // R*64
  float* a_i   = h + R * ND;                  // R*64
  float* a_j   = a_i + R * ND;                // R*64
  float* sq    = a_j + R * ND;                // R
  float* sinv  = sq + R;                      // NB*NN*NN
  float* adj   = sinv + (size_t)NB * NN * NN; // NB*NN*NN
  float* tmp   = adj + (size_t)NB * NN * NN;  // R*64
  float* h_res = tmp + R * ND;                // R*64

  proj_kernel<<<R / 4, 256, 0, stream>>>(x, proj_w, proj_b, s_w1, h, sq, a_i, a_j);
  sigma_kernel<<<dim3(NN / 16, NN / 16, NB), 256, 0, stream>>>(
      a_i, a_j, s_b1, ln_g, ln_b, s_w2, s_b2, sinv);
  gram_adj_kernel<<<dim3(2, NN / 16, NB), 256, 0, stream>>>(h, sq, sinv, adj);
  agg1_kernel<<<NB * 4, 256, 0, stream>>>(adj, h, tmp);
  agg2_kernel<<<128, 256, 0, stream>>>(tmp, gnn_w, h, h_res);
  head_kernel<<<NB, 256, 0, stream>>>(h_res, c_w1, c_b1, bn_g, bn_b, bn_mean,
                                      bn_var, c_w2, c_b2, out);
}